// ScaleAdaptiveAttention_48541720379652
// MI455X (gfx1250) — compile-verified
//
#include <hip/hip_runtime.h>

typedef float v2f __attribute__((ext_vector_type(2)));
typedef float v8f __attribute__((ext_vector_type(8)));
typedef int   v4i __attribute__((vector_size(16)));   // matches async-LDS builtin param type

#define B_   8
#define N_   32768
#define K_   64
#define D_   64
#define PITCH 68            // LDS row pitch in floats: bank = (4*row + d) % 64 -> conflict-free frags
#define LAMBDA_CONE 0.5f
#define EPS 1e-8f

#if __has_builtin(__builtin_amdgcn_global_load_async_to_lds_b128) && \
    __has_builtin(__builtin_amdgcn_s_wait_asynccnt)
#define USE_ASYNC_LDS 1
#else
#define USE_ASYNC_LDS 0
#endif

#if __has_builtin(__builtin_amdgcn_tanhf)
#define FAST_TANH(x) __builtin_amdgcn_tanhf(x)
#else
#define FAST_TANH(x) tanhf(x)
#endif

#define FAST_SQRT(x) __builtin_amdgcn_sqrtf(x)

__launch_bounds__(256)
__global__ void minkowski_attn_kernel(const float* __restrict__ features,
                                      const float* __restrict__ slots,
                                      const float* __restrict__ horizons,
                                      float* __restrict__ out)
{
    __shared__ float sSlots[K_ * PITCH];       // 17408 B
    __shared__ float sFeat[8][16 * PITCH];     // 34816 B (per-wave 16-col tiles)
    __shared__ float sSsq[K_], sSt[K_], sHinv[K_];

    const int tid   = threadIdx.x;
    const int lane  = tid & 31;
    const int wave  = tid >> 5;
    const int b     = blockIdx.y;
    const int nBase = (blockIdx.x * 8 + wave) * 16;

    const float4* gSlots = (const float4*)(slots + (size_t)b * K_ * D_);
    const float4* gFeat  = (const float4*)(features + ((size_t)b * N_ + nBase) * (size_t)D_);
    float* fW = sFeat[wave];

#if USE_ASYNC_LDS
    // ---- async DMA: global -> LDS directly (ASYNCcnt-tracked), no VGPR round trip ----
    #pragma unroll
    for (int j = 0; j < 4; ++j) {              // slots: 4096 f32, 16B per thread per step
        int i4 = j * 256 + tid;
        int e  = i4 * 4;
        int k  = e >> 6, d = e & 63;
        __builtin_amdgcn_global_load_async_to_lds_b128(
            (v4i*)(gSlots + i4), (v4i*)&sSlots[k * PITCH + d], 0, 0);
    }
    #pragma unroll
    for (int j = 0; j < 8; ++j) {              // this wave's feature tile: 1024 f32
        int i4 = j * 32 + lane;
        int e  = i4 * 4;
        int c  = e >> 6, d = e & 63;
        __builtin_amdgcn_global_load_async_to_lds_b128(
            (v4i*)(gFeat + i4), (v4i*)&fW[c * PITCH + d], 0, 0);
    }
    if (tid < K_) sHinv[tid] = 1.f / (horizons[b * K_ + tid] + EPS);
    __builtin_amdgcn_s_wait_asynccnt(0);
#else
    // ---- fallback: regular staging through VGPRs ----
    #pragma unroll
    for (int j = 0; j < 4; ++j) {
        int i4 = j * 256 + tid;
        int e  = i4 * 4;
        int k  = e >> 6, d = e & 63;
        float4 v = gSlots[i4];
        *(float4*)&sSlots[k * PITCH + d] = v;
    }
    #pragma unroll
    for (int j = 0; j < 8; ++j) {
        int i4 = j * 32 + lane;
        int e  = i4 * 4;
        int c  = e >> 6, d = e & 63;
        float4 v = gFeat[i4];
        *(float4*)&fW[c * PITCH + d] = v;
    }
    if (tid < K_) sHinv[tid] = 1.f / (horizons[b * K_ + tid] + EPS);
#endif
    __syncthreads();

    // ---- per-slot scalars: st[k] = slot time, ssq[k] = |spatial|^2 ----
    if (tid < K_) {
        const float4* row = (const float4*)&sSlots[tid * PITCH];
        float a = 0.f;
        #pragma unroll
        for (int j = 0; j < 16; ++j) {
            float4 v = row[j];
            a += v.x*v.x + v.y*v.y + v.z*v.z + v.w*v.w;
        }
        float st = sSlots[tid * PITCH];
        sSt[tid]  = st;
        sSsq[tid] = a - st * st;
    }
    __syncthreads();

    // ---- WMMA: fullcross[k,n] = sum_{d=0..63} slots[k][d]*feat[n][d] ----
    v8f acc[4] = {};                 // 4 k-tiles of 16 -> K = 64 rows
    const int lhalf = lane >> 4;     // 0: lanes 0-15, 1: lanes 16-31
    const int l15   = lane & 15;

    #pragma unroll
    for (int q = 0; q < 16; ++q) {   // K-dim steps of 4
        const int dbase = 4 * q + 2 * lhalf;   // A & B share the same lane->K mapping
        v2f bfrag;
        bfrag.x = fW[l15 * PITCH + dbase];
        bfrag.y = fW[l15 * PITCH + dbase + 1];
        #pragma unroll
        for (int t = 0; t < 4; ++t) {
            const int row = t * 16 + l15;
            v2f afrag;
            afrag.x = sSlots[row * PITCH + dbase];
            afrag.y = sSlots[row * PITCH + dbase + 1];
            acc[t] = __builtin_amdgcn_wmma_f32_16x16x4_f32(
                false, afrag, false, bfrag, (short)0, acc[t], false, false);
        }
    }

    // ---- per-column scalars: ft = feat time, fsq = |spatial|^2 ----
    float ft = fW[l15 * PITCH];
    float fsqAll = 0.f;
    {
        const float4* row = (const float4*)&fW[l15 * PITCH];
        #pragma unroll
        for (int j = 0; j < 16; ++j) {
            float4 v = row[j];
            fsqAll += v.x*v.x + v.y*v.y + v.z*v.z + v.w*v.w;
        }
    }
    const float fsq = fsqAll - ft * ft;

    // ---- logits for this lane's 32 (k,n) cells ----
    float lg[4][8];
    #pragma unroll
    for (int t = 0; t < 4; ++t) {
        #pragma unroll
        for (int r = 0; r < 8; ++r) {
            const int k = t * 16 + r + lhalf * 8;     // C-tile: M=r (lo half), M=r+8 (hi half)
            float st    = sSt[k];
            float cross = acc[t][r] - ft * st;        // remove time component from 64-d dot
            float dx2   = fmaxf(fsq - 2.f * cross + sSsq[k], 0.f);
            float dt    = ft - st;
            float itv   = dt * dt - dx2;
            float adist = FAST_SQRT(fabsf(itv) + EPS); // |sign(itv)*sqrt(|itv|+eps)|
            if (itv == 0.f) adist = 0.f;               // sign(0)=0 edge case
            float cone  = (fabsf(dt) - FAST_SQRT(dx2 + EPS)) * sHinv[k];
            lg[t][r] = -adist + LAMBDA_CONE * FAST_TANH(cone);
        }
    }

    // ---- softmax over K=64: lane L holds 32 values, lane L^16 the other 32 ----
    float m = -3.402823466e38f;
    #pragma unroll
    for (int t = 0; t < 4; ++t)
        #pragma unroll
        for (int r = 0; r < 8; ++r) m = fmaxf(m, lg[t][r]);
    m = fmaxf(m, __shfl_xor(m, 16, 32));

    float s = 0.f;
    #pragma unroll
    for (int t = 0; t < 4; ++t)
        #pragma unroll
        for (int r = 0; r < 8; ++r) { lg[t][r] = __expf(lg[t][r] - m); s += lg[t][r]; }
    s += __shfl_xor(s, 16, 32);
    const float inv = 1.f / s;

    // ---- store: out[b, k, n]; lanes write 2x64B contiguous runs per instruction ----
    float* outB = out + (size_t)b * K_ * N_ + nBase + l15;
    #pragma unroll
    for (int t = 0; t < 4; ++t)
        #pragma unroll
        for (int r = 0; r < 8; ++r) {
            const int k = t * 16 + r + lhalf * 8;
            outB[(size_t)k * N_] = lg[t][r] * inv;
        }
}

extern "C" void kernel_launch(void* const* d_in, const int* in_sizes, int n_in,
                              void* d_out, int out_size, void* d_ws, size_t ws_size,
                              hipStream_t stream) {
    const float* features = (const float*)d_in[0];   // (8, 32768, 64) f32
    const float* slots    = (const float*)d_in[1];   // (8, 64, 64)   f32
    const float* horizons = (const float*)d_in[2];   // (8, 64)       f32
    float* out = (float*)d_out;                      // (8, 64, 32768) f32

    dim3 grid(N_ / (8 * 16), B_);   // 256 x 8 blocks
    dim3 block(256);                // 8 waves (wave32) per block
    minkowski_attn_kernel<<<grid, block, 0, stream>>>(features, slots, horizons, out);
}